// MPNNEncoder_17093969838459
// MI455X (gfx1250) — compile-verified
//
#include <hip/hip_runtime.h>
#include <hip/hip_bf16.h>

typedef __attribute__((ext_vector_type(2))) float v2f;
typedef __attribute__((ext_vector_type(8))) float v8f;

// D = A(16x4 f32) * B(4x16 f32) + C(16x16 f32), full fp32 WMMA on gfx1250
__device__ __forceinline__ v8f wmma4(v2f a, v2f b, v8f c) {
    return __builtin_amdgcn_wmma_f32_16x16x4_f32(
        /*neg_a=*/false, a, /*neg_b=*/false, b,
        /*c_mod=*/(short)0, c, /*reuse_a=*/false, /*reuse_b=*/false);
}

// ---------------------------------------------------------------- embed ----
// node[n][c] = b_in[c] + sum_k classes[n][k] * W_in[c][k]   (N x 16 @ 16 x 32)
__global__ __launch_bounds__(256)
void embed_kernel(const float* __restrict__ classes, const float* __restrict__ W_in,
                  const float* __restrict__ b_in, float* __restrict__ node,
                  long long total) {
    long long gid = (long long)blockIdx.x * 256 + threadIdx.x;
    if (gid >= total) return;
    int n = (int)(gid >> 5), c = (int)(gid & 31);
    float acc = b_in[c];
#pragma unroll
    for (int k = 0; k < 16; ++k) acc += classes[n * 16 + k] * W_in[c * 16 + k];
    node[gid] = acc;
}

// ------------------------------------------------------------- edge MLP ----
// Per block: stage 64 directed edges' features x=[node[s],node[d],pos[d]-pos[s],0]
// ([64 x 68] in LDS), then 4 waves each run a 16-row tile through the 3-layer
// MLP with V_WMMA_F32_16X16X4_F32 and atomically scatter m into a[src].
#define EK_BLOCK 128
#define EK_TILE  64
#define XW       68  // 67 features + 1 zero pad (272B rows, 16B aligned)

__global__ __launch_bounds__(EK_BLOCK)
void edge_mlp_kernel(const float* __restrict__ node, const float* __restrict__ pos,
                     const int* __restrict__ edges,  // [2][E]
                     const float* __restrict__ W1, const float* __restrict__ b1,
                     const float* __restrict__ W2, const float* __restrict__ b2,
                     const float* __restrict__ W3, const float* __restrict__ b3,
                     float* __restrict__ a_out, int n_edges) {
    __shared__ __align__(16) float X[EK_TILE * XW];    // 17408 B
    __shared__ __align__(16) float W1p[34 * 32 * 2];   // 8704 B, pairs (k,k+1), B layout
    __shared__ __align__(16) float W2p[16 * 32 * 2];   // 4096 B
    __shared__ __align__(16) float W3p[16 * 32 * 2];   // 4096 B
    __shared__ __align__(16) float Y[4][16 * 32];      // per-wave layer buffer, 8192 B
    __shared__ float B1s[32], B2s[32], B3s[32];
    __shared__ int   SRC[EK_TILE];

    const int tid = threadIdx.x;
    const long long twoE = 2LL * n_edges;

    // ---- stage packed weights (B layout, float2 over K pairs) ----
    for (int i = tid; i < 34 * 32; i += EK_BLOCK) {
        int kk = i >> 5, n = i & 31;
        int k0 = 2 * kk, k1 = 2 * kk + 1;
        W1p[2 * i + 0] = (k0 < 67) ? W1[n * 67 + k0] : 0.0f;
        W1p[2 * i + 1] = (k1 < 67) ? W1[n * 67 + k1] : 0.0f;
    }
    for (int i = tid; i < 16 * 32; i += EK_BLOCK) {
        int kk = i >> 5, n = i & 31;
        W2p[2 * i + 0] = W2[n * 32 + 2 * kk];
        W2p[2 * i + 1] = W2[n * 32 + 2 * kk + 1];
        W3p[2 * i + 0] = W3[n * 32 + 2 * kk];
        W3p[2 * i + 1] = W3[n * 32 + 2 * kk + 1];
    }
    if (tid < 32) { B1s[tid] = b1[tid]; B2s[tid] = b2[tid]; B3s[tid] = b3[tid]; }

    // ---- stage 64 edge-feature rows: 2 threads per row ----
    const long long base = (long long)blockIdx.x * EK_TILE;
    {
        int r = tid >> 1, half = tid & 1;
        long long e = base + r;
        if (e >= twoE) e = 0;                      // guard (2E is an exact multiple anyway)
        int s, d;
        if (e < n_edges) { s = edges[e]; d = edges[n_edges + e]; }
        else             { long long e2 = e - n_edges; s = edges[n_edges + e2]; d = edges[e2]; }
        const float4* srow = (const float4*)(node + (long long)(half ? d : s) * 32);
        float4* xrow = (float4*)(&X[r * XW + half * 32]);
#pragma unroll
        for (int q = 0; q < 8; ++q) xrow[q] = srow[q];
        if (half) {
            X[r * XW + 64] = pos[3 * d + 0] - pos[3 * s + 0];
            X[r * XW + 65] = pos[3 * d + 1] - pos[3 * s + 1];
            X[r * XW + 66] = pos[3 * d + 2] - pos[3 * s + 2];
            X[r * XW + 67] = 0.0f;
            SRC[r] = s;
        }
    }
    __syncthreads();

    // ---- per-wave 16-edge tile through the MLP ----
    const int wave = tid >> 5, lane = tid & 31;
    const int hi = lane >> 4, col = lane & 15;
    const int row0 = wave * 16;
    float* Yw = Y[wave];
    const v2f* Arow1 = (const v2f*)&X[(row0 + col) * XW];  // lane's A row (M = lane%16)
    const v2f* W1v = (const v2f*)W1p;
    const v2f* W2v = (const v2f*)W2p;
    const v2f* W3v = (const v2f*)W3p;

    // Layer 1: K = 68 (17 k-steps), N = 32 (2 tiles)
    v8f acc0 = {}, acc1 = {};
#pragma unroll
    for (int kk = 0; kk < 17; ++kk) {
        v2f af = Arow1[2 * kk + hi];
        v2f b0 = W1v[(2 * kk + hi) * 32 + col];
        v2f b1f = W1v[(2 * kk + hi) * 32 + col + 16];
        acc0 = wmma4(af, b0, acc0);
        acc1 = wmma4(af, b1f, acc1);
    }
    {   // bias + relu, C layout -> LDS (A layout for next layer)
        float bb0 = B1s[col], bb1 = B1s[col + 16];
#pragma unroll
        for (int j = 0; j < 8; ++j) {
            int m = j + 8 * hi;
            Yw[m * 32 + col]      = fmaxf(acc0[j] + bb0, 0.0f);
            Yw[m * 32 + 16 + col] = fmaxf(acc1[j] + bb1, 0.0f);
        }
    }

    // Layer 2: K = 32 (8 k-steps)
    const v2f* Arow2 = (const v2f*)&Yw[col * 32];
    v8f c20 = {}, c21 = {};
#pragma unroll
    for (int kk = 0; kk < 8; ++kk) {
        v2f af = Arow2[2 * kk + hi];
        c20 = wmma4(af, W2v[(2 * kk + hi) * 32 + col], c20);
        c21 = wmma4(af, W2v[(2 * kk + hi) * 32 + col + 16], c21);
    }
    {
        float bb0 = B2s[col], bb1 = B2s[col + 16];
#pragma unroll
        for (int j = 0; j < 8; ++j) {
            int m = j + 8 * hi;
            Yw[m * 32 + col]      = fmaxf(c20[j] + bb0, 0.0f);
            Yw[m * 32 + 16 + col] = fmaxf(c21[j] + bb1, 0.0f);
        }
    }

    // Layer 3: K = 32
    v8f c30 = {}, c31 = {};
#pragma unroll
    for (int kk = 0; kk < 8; ++kk) {
        v2f af = Arow2[2 * kk + hi];
        c30 = wmma4(af, W3v[(2 * kk + hi) * 32 + col], c30);
        c31 = wmma4(af, W3v[(2 * kk + hi) * 32 + col + 16], c31);
    }

    // segment-sum scatter: a[src] += m   (16 lanes -> 64B contiguous per row half)
    float bb0 = B3s[col], bb1 = B3s[col + 16];
#pragma unroll
    for (int j = 0; j < 8; ++j) {
        int m = j + 8 * hi;
        long long s = SRC[row0 + m];
        atomicAdd(&a_out[s * 32 + col],      c30[j] + bb0);
        atomicAdd(&a_out[s * 32 + 16 + col], c31[j] + bb1);
    }
}

// ------------------------------------------------------------- node GRU ----
// gi = a@Wih^T + bih, gh = h@Whh^T + bhh  (both [16,32]@[32,96] per wave-tile)
// r,z = sigmoid; n = tanh(i_n + r*h_n); h' = (1-z)*n + z*h. Zeroes `a` after use.
#define GK_BLOCK 128

__global__ __launch_bounds__(GK_BLOCK)
void node_gru_kernel(float* __restrict__ a, float* __restrict__ node,
                     const float* __restrict__ Wih, const float* __restrict__ Whh,
                     const float* __restrict__ bih, const float* __restrict__ bhh,
                     int n_nodes) {
    __shared__ __align__(16) float Wihp[16 * 96 * 2];  // 12 KB, pair-packed B layout
    __shared__ __align__(16) float Whhp[16 * 96 * 2];  // 12 KB
    __shared__ float BI[96], BH[96];

    const int tid = threadIdx.x;
    for (int i = tid; i < 16 * 96; i += GK_BLOCK) {
        int kk = i / 96, n = i % 96;
        Wihp[2 * i + 0] = Wih[n * 32 + 2 * kk];
        Wihp[2 * i + 1] = Wih[n * 32 + 2 * kk + 1];
        Whhp[2 * i + 0] = Whh[n * 32 + 2 * kk];
        Whhp[2 * i + 1] = Whh[n * 32 + 2 * kk + 1];
    }
    if (tid < 96) { BI[tid] = bih[tid]; BH[tid] = bhh[tid]; }
    __syncthreads();

    const int wave = tid >> 5, lane = tid & 31;
    const int hi = lane >> 4, col = lane & 15;
    long long tile = (long long)blockIdx.x * (GK_BLOCK / 32) + wave;
    if (tile * 16 >= n_nodes) return;               // uniform per wave
    const long long rbase = tile * 16;

    const v2f* arow = (const v2f*)(a + (rbase + col) * 32);
    const v2f* hrow = (const v2f*)(node + (rbase + col) * 32);
    const v2f* Wiv = (const v2f*)Wihp;
    const v2f* Whv = (const v2f*)Whhp;

    v8f gi[6], gh[6];
#pragma unroll
    for (int t = 0; t < 6; ++t) { gi[t] = {}; gh[t] = {}; }
#pragma unroll
    for (int kk = 0; kk < 8; ++kk) {
        v2f af = arow[2 * kk + hi];
        v2f hf = hrow[2 * kk + hi];
#pragma unroll
        for (int nt = 0; nt < 6; ++nt) {
            gi[nt] = wmma4(af, Wiv[(2 * kk + hi) * 96 + nt * 16 + col], gi[nt]);
            gh[nt] = wmma4(hf, Whv[(2 * kk + hi) * 96 + nt * 16 + col], gh[nt]);
        }
    }

#pragma unroll
    for (int j = 0; j < 8; ++j) {
        int m = j + 8 * hi;
        long long r = rbase + m;
#pragma unroll
        for (int half = 0; half < 2; ++half) {
            int c = half * 16 + col;  // output column 0..31
            float i_r = gi[half][j]     + BI[c];
            float i_z = gi[2 + half][j] + BI[32 + c];
            float i_n = gi[4 + half][j] + BI[64 + c];
            float h_r = gh[half][j]     + BH[c];
            float h_z = gh[2 + half][j] + BH[32 + c];
            float h_n = gh[4 + half][j] + BH[64 + c];
            float rg = 1.0f / (1.0f + __expf(-(i_r + h_r)));
            float zg = 1.0f / (1.0f + __expf(-(i_z + h_z)));
            float ng = tanhf(i_n + rg * h_n);
            float hold = node[r * 32 + c];
            node[r * 32 + c] = (1.0f - zg) * ng + zg * hold;
            a[r * 32 + c] = 0.0f;   // re-zero accumulator for next iteration
        }
    }
}

// ---------------------------------------------------------------- launch ---
extern "C" void kernel_launch(void* const* d_in, const int* in_sizes, int n_in,
                              void* d_out, int out_size, void* d_ws, size_t ws_size,
                              hipStream_t stream) {
    const float* pos     = (const float*)d_in[0];
    const float* classes = (const float*)d_in[1];
    const int*   edges   = (const int*)  d_in[2];
    // d_in[3] = n_graphs (unused: output is just node, flat)
    const float* W_in = (const float*)d_in[4];
    const float* b_in = (const float*)d_in[5];
    const float* W1 = (const float*)d_in[6],  *b1 = (const float*)d_in[7];
    const float* W2 = (const float*)d_in[8],  *b2 = (const float*)d_in[9];
    const float* W3 = (const float*)d_in[10], *b3 = (const float*)d_in[11];
    const float* Wih = (const float*)d_in[12], *Whh = (const float*)d_in[13];
    const float* bih = (const float*)d_in[14], *bhh = (const float*)d_in[15];
    // d_in[16..19]: edge GRU params — dead code in the reference (edge_feat unused)

    const int n_nodes = in_sizes[1] / 16;   // classes is [N,16]
    const int n_edges = in_sizes[2] / 2;    // edges is [2,E]
    const long long nodef = (long long)n_nodes * 32;

    float* node = (float*)d_ws;
    float* a    = node + nodef;

    hipMemsetAsync(a, 0, nodef * sizeof(float), stream);

    embed_kernel<<<(int)((nodef + 255) / 256), 256, 0, stream>>>(classes, W_in, b_in, node, nodef);

    const int eblocks = (int)((2LL * n_edges + EK_TILE - 1) / EK_TILE);
    const int tiles   = (n_nodes + 15) / 16;
    const int gblocks = (tiles + (GK_BLOCK / 32) - 1) / (GK_BLOCK / 32);

    for (int it = 0; it < 6; ++it) {
        edge_mlp_kernel<<<eblocks, EK_BLOCK, 0, stream>>>(
            node, pos, edges, W1, b1, W2, b2, W3, b3, a, n_edges);
        node_gru_kernel<<<gblocks, GK_BLOCK, 0, stream>>>(
            a, node, Wih, Whh, bih, bhh, n_nodes);
    }

    hipMemcpyAsync(d_out, node, nodef * sizeof(float), hipMemcpyDeviceToDevice, stream);
}